// Protonet_49306224558301
// MI455X (gfx1250) — compile-verified
//
#include <hip/hip_runtime.h>

typedef __attribute__((ext_vector_type(2))) float v2f;
typedef __attribute__((ext_vector_type(8))) float v8f;

// ---------------------------------------------------------------------------
// Problem constants (from reference setup)
// ---------------------------------------------------------------------------
#define N_CLASS   256
#define N_SUPPORT 5
#define N_QUERY   15
#define X_DIM     2048
#define Z_DIM     128
#define N_COARSE  8
#define N_SUP_ROWS (N_CLASS * N_SUPPORT)   // 1280
#define N_QRY_ROWS (N_CLASS * N_QUERY)     // 3840

// ---------------------------------------------------------------------------
// Kernel 1: Wsum[d,z] = sum_k W_fine[k,d,z]      (2048*128 elements)
// ---------------------------------------------------------------------------
__global__ void wsum_kernel(const float* __restrict__ W_fine, float* __restrict__ wsum) {
    int idx = blockIdx.x * 256 + threadIdx.x;
    if (idx < X_DIM * Z_DIM) {
        float s = 0.f;
        #pragma unroll
        for (int k = 0; k < N_COARSE; ++k)
            s += W_fine[(size_t)k * X_DIM * Z_DIM + idx];
        wsum[idx] = s;
    }
}

// ---------------------------------------------------------------------------
// Kernel 2: b_comb[z] = sum_k b_fine[k,z] + sum_d b_share[d]*Wsum[d,z]
// ---------------------------------------------------------------------------
__global__ void bcomb_kernel(const float* __restrict__ b_share,
                             const float* __restrict__ b_fine,
                             const float* __restrict__ wsum,
                             float* __restrict__ b_comb) {
    int zd = threadIdx.x;   // 128 threads
    float s = 0.f;
    #pragma unroll
    for (int k = 0; k < N_COARSE; ++k) s += b_fine[k * Z_DIM + zd];
    float t = 0.f;
    for (int d = 0; d < X_DIM; ++d) t += b_share[d] * wsum[(size_t)d * Z_DIM + zd];
    b_comb[zd] = s + t;
}

// ---------------------------------------------------------------------------
// WMMA f32 GEMM: C[M,N] = A[M,K] @ B[K,N] (+ bias[N] if non-null)
// One 16x16 output tile per wave, V_WMMA_F32_16X16X4_F32, K-loop step 4.
// A-layout (ISA 7.12.2, 32-bit A 16x4): lane = half*16 + lo,
//   lane holds A[row0+lo][k + 2*half .. +1]  -> one float2 load.
// B-layout (32-bit, K-split mirrored):  lane holds B[k+2*half..+1][col0+lo].
// C/D layout: VGPR r holds row (row0 + r + 8*half), col (col0 + lo).
// ---------------------------------------------------------------------------
__global__ void __launch_bounds__(256)
gemm_wmma_f32(const float* __restrict__ A, const float* __restrict__ B,
              const float* __restrict__ bias, float* __restrict__ C,
              int M, int N, int K) {
    const int wave = threadIdx.x >> 5;
    const int lane = threadIdx.x & 31;
    const int tilesN = N >> 4;
    const int totalTiles = (M >> 4) * tilesN;
    const int tile = blockIdx.x * 8 + wave;
    if (tile >= totalTiles) return;          // wave-uniform: EXEC stays all-ones

    const int row0 = (tile / tilesN) << 4;
    const int col0 = (tile % tilesN) << 4;
    const int lo   = lane & 15;
    const int half = lane >> 4;

    const float* Arow = A + (size_t)(row0 + lo) * K + 2 * half;   // 8B-aligned
    const float* Bcol = B + (size_t)(2 * half) * N + col0 + lo;

    v8f acc = {};
    #pragma unroll 4
    for (int k = 0; k < K; k += 4) {
        v2f a = *(const v2f*)(Arow + k);
        v2f b;
        b.x = Bcol[(size_t)k * N];
        b.y = Bcol[(size_t)(k + 1) * N];
        acc = __builtin_amdgcn_wmma_f32_16x16x4_f32(
            /*neg_a=*/false, a, /*neg_b=*/false, b,
            /*c_mod=*/(short)0, acc, /*reuse_a=*/false, /*reuse_b=*/false);
    }

    const float bv = bias ? bias[col0 + lo] : 0.f;
    #pragma unroll
    for (int r = 0; r < 8; ++r) {
        const int row = row0 + r + half * 8;
        C[(size_t)row * N + col0 + lo] = acc[r] + bv;
    }
}

// ---------------------------------------------------------------------------
// Kernel 5: prototypes = mean over 5 support embeddings per class
// ---------------------------------------------------------------------------
__global__ void proto_kernel(const float* __restrict__ z, float* __restrict__ proto) {
    int idx = blockIdx.x * 256 + threadIdx.x;      // 256*128 = 32768
    if (idx < N_CLASS * Z_DIM) {
        int c = idx >> 7, d = idx & 127;
        const float* base = z + (size_t)c * N_SUPPORT * Z_DIM + d;
        float s = base[0] + base[Z_DIM] + base[2 * Z_DIM] + base[3 * Z_DIM] + base[4 * Z_DIM];
        proto[idx] = s * 0.2f;
    }
}

// ---------------------------------------------------------------------------
// Kernel 6: per-query dists -> log-softmax target term + argmin correctness
// One block (256 threads) per query; thread t owns class t.
// ---------------------------------------------------------------------------
__global__ void __launch_bounds__(256)
query_kernel(const float* __restrict__ z, const float* __restrict__ proto,
             float* __restrict__ loss_pq, float* __restrict__ acc_pq) {
    const int q = blockIdx.x;        // 0..3839
    const int t = threadIdx.x;       // 0..255 = class id

    __shared__ float s_zq[Z_DIM];
    __shared__ float s_dist[N_CLASS];

    const float* zq = z + (size_t)(N_SUP_ROWS + q) * Z_DIM;
    if (t < Z_DIM) s_zq[t] = zq[t];
    __syncthreads();

    const float* p = proto + (size_t)t * Z_DIM;
    float d = 0.f;
    #pragma unroll 8
    for (int i = 0; i < Z_DIM; ++i) {
        float df = s_zq[i] - p[i];
        d += df * df;
    }
    s_dist[t] = d;
    __syncthreads();

    if (t == 0) {
        const int cq = q / N_QUERY;
        float dmin = s_dist[0];
        int amin = 0;
        for (int c = 1; c < N_CLASS; ++c)
            if (s_dist[c] < dmin) { dmin = s_dist[c]; amin = c; }   // first-min == jnp.argmax
        float se = 0.f;
        for (int c = 0; c < N_CLASS; ++c) se += expf(dmin - s_dist[c]);
        const float lse = logf(se) - dmin;          // logsumexp_c(-dist)
        loss_pq[q] = s_dist[cq] + lse;              // -log_p_y[target]
        acc_pq[q]  = (amin == cq) ? 1.f : 0.f;
    }
}

// ---------------------------------------------------------------------------
// Kernel 7: deterministic fixed-order reduction -> d_out = {loss, acc}
// ---------------------------------------------------------------------------
__global__ void __launch_bounds__(256)
reduce_kernel(const float* __restrict__ loss_pq, const float* __restrict__ acc_pq,
              float* __restrict__ out) {
    __shared__ float sl[256];
    __shared__ float sa[256];
    const int t = threadIdx.x;
    float l = 0.f, a = 0.f;
    for (int i = t; i < N_QRY_ROWS; i += 256) { l += loss_pq[i]; a += acc_pq[i]; }
    sl[t] = l; sa[t] = a;
    __syncthreads();
    for (int s = 128; s > 0; s >>= 1) {
        if (t < s) { sl[t] += sl[t + s]; sa[t] += sa[t + s]; }
        __syncthreads();
    }
    if (t == 0) {
        out[0] = sl[0] / (float)N_QRY_ROWS;
        out[1] = sa[0] / (float)N_QRY_ROWS;
    }
}

// ---------------------------------------------------------------------------
// Host-side orchestration
// ---------------------------------------------------------------------------
extern "C" void kernel_launch(void* const* d_in, const int* in_sizes, int n_in,
                              void* d_out, int out_size, void* d_ws, size_t ws_size,
                              hipStream_t stream) {
    const float* xs      = (const float*)d_in[0];  // [256,5,2048]
    const float* xq      = (const float*)d_in[1];  // [256,15,2048]
    const float* W_share = (const float*)d_in[2];  // [2048,2048]
    const float* b_share = (const float*)d_in[3];  // [2048]
    // d_in[4] W_coarse, d_in[5] b_coarse: dead code in the reference output
    const float* W_fine  = (const float*)d_in[6];  // [8,2048,128]
    const float* b_fine  = (const float*)d_in[7];  // [8,128]

    float* ws      = (float*)d_ws;
    float* wsum    = ws;                                   // 2048*128
    float* wcomb   = wsum    + X_DIM * Z_DIM;              // 2048*128
    float* bcomb   = wcomb   + X_DIM * Z_DIM;              // 128
    float* zbuf    = bcomb   + Z_DIM;                      // 5120*128
    float* proto   = zbuf    + (N_SUP_ROWS + N_QRY_ROWS) * Z_DIM;  // 256*128
    float* loss_pq = proto   + N_CLASS * Z_DIM;            // 3840
    float* acc_pq  = loss_pq + N_QRY_ROWS;                 // 3840

    // 1. Wsum = sum_k W_fine[k]
    wsum_kernel<<<(X_DIM * Z_DIM + 255) / 256, 256, 0, stream>>>(W_fine, wsum);

    // 2. combined bias
    bcomb_kernel<<<1, Z_DIM, 0, stream>>>(b_share, b_fine, wsum, bcomb);

    // 3. W_comb = W_share @ Wsum        (M=2048, N=128, K=2048) -> 1024 tiles
    gemm_wmma_f32<<<(2048 / 16) * (128 / 16) / 8, 256, 0, stream>>>(
        W_share, wsum, nullptr, wcomb, 2048, 128, 2048);

    // 4a. z[0:1280]   = xs @ W_comb + b_comb   (M=1280) -> 640 tiles
    gemm_wmma_f32<<<(1280 / 16) * (128 / 16) / 8, 256, 0, stream>>>(
        xs, wcomb, bcomb, zbuf, 1280, 128, 2048);
    // 4b. z[1280:5120] = xq @ W_comb + b_comb  (M=3840) -> 1920 tiles
    gemm_wmma_f32<<<(3840 / 16) * (128 / 16) / 8, 256, 0, stream>>>(
        xq, wcomb, bcomb, zbuf + (size_t)N_SUP_ROWS * Z_DIM, 3840, 128, 2048);

    // 5. class prototypes
    proto_kernel<<<(N_CLASS * Z_DIM + 255) / 256, 256, 0, stream>>>(zbuf, proto);

    // 6. per-query loss / correctness
    query_kernel<<<N_QRY_ROWS, 256, 0, stream>>>(zbuf, proto, loss_pq, acc_pq);

    // 7. deterministic reduction to the two scalars
    reduce_kernel<<<1, 256, 0, stream>>>(loss_pq, acc_pq, (float*)d_out);
}